// CLIPVisionTower_GOPrune_52003464020299
// MI455X (gfx1250) — compile-verified
//
#include <hip/hip_runtime.h>
#include <hip/hip_bf16.h>

// ---------------------------------------------------------------------------
// Problem constants
// ---------------------------------------------------------------------------
#define NROWS   73856      // 8*16*577 rows of attn, each 577 floats
#define ROWLEN  577
#define NCOLS   576        // score columns (attn cols 1..576)
#define NTILES  36         // 576 / 16
#define YBLKS   64         // row-split blocks
#define NTOK    288        // output rows
#define HDIM    1024

typedef float v2f __attribute__((ext_vector_type(2)));
typedef float v8f __attribute__((ext_vector_type(8)));

// ---------------------------------------------------------------------------
// Kernel 1: column-sum of attn via V_WMMA_F32_16X16X4_F32 with all-ones A.
//   D[m][n] = sum_k 1 * B[k][n] + C[m][n]  -> every D row = 4-row column sum.
// Grid: (36 column tiles, 64 row splits) x 256 threads (8 waves).
// Wave wy = blockIdx.y*8+wave handles 8-row chunks {wy*8 + 4096*i} as two
// independent 4-row WMMA accumulator chains (c0, c1) so loads of one group
// can overlap the other group's WMMA. 73856 % 8 == 0 -> exact coverage,
// each row read exactly once. Deterministic: fixed-order combines, no atomics.
// ---------------------------------------------------------------------------
__global__ __launch_bounds__(256) void colsum_wmma_kernel(
    const float* __restrict__ attn, float* __restrict__ partial) {
  const int tile   = blockIdx.x;            // 16-column tile
  const int waveId = threadIdx.x >> 5;
  const int lane   = threadIdx.x & 31;
  const int half   = lane >> 4;             // B row-half (lanes 16-31 = rows 2,3)
  const int n      = lane & 15;             // column within tile
  const int col    = 1 + tile * 16 + n;     // attn column (skip CLS col 0)
  const int wy     = blockIdx.y * 8 + waveId;   // 0..511

  v8f c0 = {0.f, 0.f, 0.f, 0.f, 0.f, 0.f, 0.f, 0.f};
  v8f c1 = {0.f, 0.f, 0.f, 0.f, 0.f, 0.f, 0.f, 0.f};
  v2f a; a.x = 1.0f; a.y = 1.0f;            // all-ones A (layout-invariant)

  // Per-lane base pointer for this lane's (column, row-half) stream.
  const float* base = attn + (size_t)col + (size_t)(wy * 8 + 2 * half) * ROWLEN;
  const size_t stepRow  = (size_t)ROWLEN;           // +1 row
  const size_t stepIter = (size_t)4096 * ROWLEN;    // +4096 rows per iteration

  for (int r = wy * 8; r < NROWS; r += 4096) {
    v2f b0, b1;
    b0.x = base[0];                 // group0: rows r+2*half, r+2*half+1
    b0.y = base[stepRow];
    b1.x = base[4 * stepRow];       // group1: rows r+4+2*half, r+4+2*half+1
    b1.y = base[5 * stepRow];
    c0 = __builtin_amdgcn_wmma_f32_16x16x4_f32(
        false, a, false, b0, (short)0, c0, false, false);
    c1 = __builtin_amdgcn_wmma_f32_16x16x4_f32(
        false, a, false, b1, (short)0, c1, false, false);
    base += stepIter;
  }

  // D row 0: lane L (0..15) holds colsum for column n=L (both chains).
  __shared__ float lds[8 * 16];
  if (lane < 16) lds[waveId * 16 + n] = c0[0] + c1[0];
  __syncthreads();
  if (threadIdx.x < 16) {
    float s = 0.f;
    for (int w = 0; w < 8; ++w) s += lds[w * 16 + threadIdx.x];  // fixed order
    partial[blockIdx.y * NCOLS + tile * 16 + threadIdx.x] = s;
  }
}

// ---------------------------------------------------------------------------
// Kernel 2: single block, 576 threads. Finalize scores (double accumulation
// over 64 partials, fixed order), per-quadrant top-36 by exact rank counting,
// 2x2 window argmax with first-index tie-break, then ascending enumeration of
// the (distinct) valid positions == argsort-by-position of the reference.
// posOut[i] = grid position (0..575) for output row i, or -1 if invalid.
// ---------------------------------------------------------------------------
__global__ __launch_bounds__(576) void select_kernel(
    const float* __restrict__ partial, int* __restrict__ posOut) {
  __shared__ float sc[NCOLS];       // score per 24x24 grid position
  __shared__ int   sel[NCOLS];      // top-k mask per grid position
  __shared__ int   flagged[NCOLS];  // valid-output mask per grid position
  __shared__ int   nvalid;

  const int t = threadIdx.x;

  // --- scores (deterministic: fixed-order double sum over 64 partials) ---
  double tot = 0.0;
  for (int y = 0; y < YBLKS; ++y) tot += (double)partial[y * NCOLS + t];
  sc[t] = (float)(tot / (128.0 * 577.0));   // mean over (8,16) then axis 0
  flagged[t] = 0;
  __syncthreads();

  // --- top-36 per quadrant: rank = #{(v2>v) || (v2==v && j2<j)} < 36 ---
  const int q  = t / 144, j  = t % 144;
  const int qr = q >> 1,  qc = q & 1;
  const int lr = j / 12,  lc = j % 12;
  const int p  = (qr * 12 + lr) * 24 + (qc * 12 + lc);
  const float v = sc[p];
  int cnt = 0;
  for (int j2 = 0; j2 < 144; ++j2) {
    const int p2 = (qr * 12 + j2 / 12) * 24 + (qc * 12 + j2 % 12);
    const float v2 = sc[p2];
    cnt += (v2 > v) || (v2 == v && j2 < j);
  }
  const int isTop = (cnt < 36);
  sel[p] = isTop;
  if (isTop) flagged[p] = 1;        // t -> p is a bijection: no races
  __syncthreads();

  // --- 2x2 window argmax over masked grid (first max wins, jnp.argmax) ---
  if (t < 144) {
    const int wq = t / 36, w = t % 36;
    const int wqr = wq >> 1, wqc = wq & 1;
    const int wr = w / 6, wc = w % 6;
    float mx = -3.0e38f; int idx = 0;
    for (int e = 0; e < 4; ++e) {
      const int rr = wqr * 12 + 2 * wr + (e >> 1);
      const int cc = wqc * 12 + 2 * wc + (e & 1);
      const int pp = rr * 24 + cc;
      const float val = sel[pp] ? -1.0e30f : sc[pp];
      if (val > mx) { mx = val; idx = e; }
    }
    if (mx > -5.0e29f) {            // valid window (not fully masked)
      const int fr = wqr * 12 + 2 * wr + (idx >> 1);
      const int fc = wqc * 12 + 2 * wc + (idx & 1);
      flagged[fr * 24 + fc] = 1;    // distinct from all other valid pos
    }
  }
  __syncthreads();

  // --- ascending enumeration of flagged == stable argsort by position ---
  if (t == 0) {
    int s = 0;
    for (int i = 0; i < NCOLS; ++i) s += flagged[i];
    nvalid = s;
  }
  const int isF = flagged[t];
  int rank = 0;
  if (isF) for (int i = 0; i < t; ++i) rank += flagged[i];
  __syncthreads();
  if (isF) posOut[rank] = t;
  if (t < NTOK && t >= nvalid) posOut[t] = -1;
}

// ---------------------------------------------------------------------------
// Kernel 3: gather features (float4) + valid flags.
// Grid: 288 blocks x 256 threads; each block copies one 1024-float row.
// ---------------------------------------------------------------------------
__global__ __launch_bounds__(256) void gather_kernel(
    const float* __restrict__ hidden, const int* __restrict__ posOut,
    float* __restrict__ out) {
  const int row = blockIdx.x;
  const int pos = posOut[row];
  float4* dst = (float4*)(out + (size_t)row * HDIM);
  if (pos >= 0) {
    const float4* src = (const float4*)(hidden + (size_t)(pos + 1) * HDIM);
    dst[threadIdx.x] = src[threadIdx.x];
  } else {
    float4 z; z.x = 0.f; z.y = 0.f; z.z = 0.f; z.w = 0.f;
    dst[threadIdx.x] = z;
  }
  if (threadIdx.x == 0)
    out[(size_t)NTOK * HDIM + row] = (pos >= 0) ? 1.0f : 0.0f;
}

// ---------------------------------------------------------------------------
extern "C" void kernel_launch(void* const* d_in, const int* in_sizes, int n_in,
                              void* d_out, int out_size, void* d_ws, size_t ws_size,
                              hipStream_t stream) {
  const float* attn   = (const float*)d_in[0];   // (8,16,577,577) f32
  const float* hidden = (const float*)d_in[1];   // (1,577,1024)   f32
  float* out = (float*)d_out;                    // 288*1024 feats + 288 flags

  float* partial = (float*)d_ws;                                  // 64*576 f32
  int*   posOut  = (int*)((char*)d_ws + (size_t)YBLKS * NCOLS * sizeof(float));

  colsum_wmma_kernel<<<dim3(NTILES, YBLKS), 256, 0, stream>>>(attn, partial);
  select_kernel<<<1, NCOLS, 0, stream>>>(partial, posOut);
  gather_kernel<<<NTOK, 256, 0, stream>>>(hidden, posOut, out);
}